// LNClampPowerScaleMap_9663676416219
// MI455X (gfx1250) — compile-verified
//
#include <hip/hip_runtime.h>
#include <hip/hip_bf16.h>
#include <math.h>

// Row-wise top-k(250) normalize + exact-GELU + scale for [2048, 21841] fp32.
// One workgroup per row; whole row staged in LDS (87.4 KB of the 320 KB WGP
// LDS) via the CDNA5 Tensor Data Mover (tensor_load_to_lds + s_wait_tensorcnt),
// exact MSB-first radix-select for the 250th-largest value using per-wave
// privatized LDS histograms (kills same-address LDS atomic serialization on
// the ~14 hot exponent bins of N(0,16) data), fp64 accumulation (reference
// computes in float64), and a final elementwise map. HBM traffic = 1 read +
// 1 write per element (roofline ~15 us at 23.3 TB/s).

#define K_TOP      250
#define V_CLASSES  21841
#define BLOCK_T    512
#define NWAVE      (BLOCK_T / 32)
#define NCOPY      NWAVE            // one histogram copy per wave32
#define HSTRIDE    260              // 256 bins + 4 pad words -> copies spread across banks
#define CLAMP_LIMF 1e15f
#define VAR_EPS    1e-8
#define OUT_SCALE  0.4

typedef __attribute__((ext_vector_type(4))) unsigned int v4u;
typedef __attribute__((ext_vector_type(8))) int          v8i;
typedef __attribute__((ext_vector_type(4))) int          v4i;

__device__ __forceinline__ float clampf(float x) {
    return fminf(fmaxf(x, -CLAMP_LIMF), CLAMP_LIMF);
}

// Monotonic bijection: float order -> unsigned order (larger float => larger key)
__device__ __forceinline__ unsigned key_of(float f) {
    unsigned u = __float_as_uint(f);
    return (u & 0x80000000u) ? ~u : (u | 0x80000000u);
}
__device__ __forceinline__ float val_of(unsigned k) {
    unsigned u = (k & 0x80000000u) ? (k & 0x7fffffffu) : ~k;
    return __uint_as_float(u);
}

#if __has_builtin(__builtin_amdgcn_tensor_load_to_lds) && __has_builtin(__builtin_amdgcn_s_wait_tensorcnt)
#define USE_TDM 1
#endif

__global__ __launch_bounds__(BLOCK_T)
void LNClampPowerScaleMap_kernel(const float* __restrict__ in,
                                 float* __restrict__ out,
                                 int V)
{
    extern __shared__ __attribute__((aligned(16))) unsigned char smem_raw[];
    float*  vals  = (float*)smem_raw;               // V floats (row staged in LDS)
    float4* vals4 = (float4*)smem_raw;              // 16B-aligned chunked view

    __shared__ unsigned whist[NCOPY * HSTRIDE];     // per-wave privatized histograms
    __shared__ unsigned hist[256];                  // reduced histogram
    __shared__ double   red_s[NWAVE];
    __shared__ double   red_s2[NWAVE];
    __shared__ unsigned bc_prefix, bc_krem;
    __shared__ double   bc_mean, bc_rsig;

    const int tid    = threadIdx.x;
    const int nt     = blockDim.x;
    const int row    = blockIdx.x;
    const int mywave = tid >> 5;
    unsigned* myhist = &whist[(mywave & (NCOPY - 1)) * HSTRIDE];
    const float* __restrict__ rowp = in + (size_t)row * (size_t)V;

    const int nchunk = V >> 2;          // 5460 float4 chunks
    const int tail0  = nchunk << 2;     // first tail element (21840)

    // ---------------- Stage row into LDS (TDM async DMA) ----------------
#ifdef USE_TDM
    if (tid < 32) {  // wave 0 only: TDM issues once per wave (EXEC ignored)
        unsigned           lds_off = (unsigned)(size_t)(void*)vals;  // low 32b of generic addr = LDS offset
        unsigned long long ga      = (unsigned long long)(const void*)rowp;
        const unsigned uV = (unsigned)V;
        // D# group 0: count=1 | lds_addr | global_addr | type=2
        v4u g0 = { 1u,
                   lds_off,
                   (unsigned)(ga & 0xffffffffu),
                   (unsigned)((ga >> 32) & 0x01ffffffu) | 0x80000000u };
        // D# group 1: data_size=4B, tensor_dim0=V, tensor_dim1=1,
        //             tile_dim0=V, tile_dim1=1, tensor_dim0_stride=V
        v8i g1 = { (int)(2u << 16),
                   (int)((uV & 0xffffu) << 16),
                   (int)(((uV >> 16) & 0xffffu) | (1u << 16)),
                   (int)((uV & 0xffffu) << 16),
                   1,
                   (int)uV,
                   (int)((uV & 0xffffu) << 16),  // stride0 hi16 (=0) | stride1 lo16
                   0 };
        v4i g2 = { 0, 0, 0, 0 };
        v4i g3 = { 0, 0, 0, 0 };
#if __clang_major__ >= 23
        v8i g4 = { 0, 0, 0, 0, 0, 0, 0, 0 };
        __builtin_amdgcn_tensor_load_to_lds(g0, g1, g2, g3, g4, 0);
#else
        __builtin_amdgcn_tensor_load_to_lds(g0, g1, g2, g3, 0);
#endif
        __builtin_amdgcn_s_wait_tensorcnt(0);
    }
    __syncthreads();
#else
    for (int i = tid; i < V; i += nt) vals[i] = rowp[i];
    __syncthreads();
#endif

    // ---------------- Exact radix-select: 250th-largest key ----------------
    unsigned prefix = 0u, krem = K_TOP;
    for (int shift = 24; shift >= 0; shift -= 8) {
        for (int i = tid; i < NCOPY * HSTRIDE; i += nt) whist[i] = 0u;
        __syncthreads();
        // build: privatized per-wave histograms, float4 LDS sweeps
        for (int c = tid; c < nchunk; c += nt) {
            float4 q = vals4[c];
            float fa[4] = { q.x, q.y, q.z, q.w };
#pragma unroll
            for (int j = 0; j < 4; ++j) {
                unsigned k = key_of(clampf(fa[j]));
                if (shift == 24 || (k >> (shift + 8)) == prefix)
                    atomicAdd(&myhist[(k >> shift) & 0xffu], 1u);
            }
        }
        for (int i = tail0 + tid; i < V; i += nt) {
            unsigned k = key_of(clampf(vals[i]));
            if (shift == 24 || (k >> (shift + 8)) == prefix)
                atomicAdd(&myhist[(k >> shift) & 0xffu], 1u);
        }
        __syncthreads();
        // reduce copies: 256 threads, one bin each
        for (int b = tid; b < 256; b += nt) {
            unsigned t = 0;
#pragma unroll
            for (int w = 0; w < NCOPY; ++w) t += whist[w * HSTRIDE + b];
            hist[b] = t;
        }
        __syncthreads();
        // scan from the top bin down (256 SALU steps, negligible)
        if (tid == 0) {
            unsigned c = 0; int b = 255;
            for (; b > 0; --b) {
                unsigned h = hist[b];
                if (c + h >= krem) break;
                c += h;
            }
            bc_prefix = (prefix << 8) | (unsigned)b;
            bc_krem   = krem - c;
        }
        __syncthreads();
        prefix = bc_prefix;
        krem   = bc_krem;
    }
    const unsigned Kstar   = prefix;  // exact 250th-largest key
    const unsigned tieNeed = krem;    // # elements equal to Kstar inside top-k

    // ---------------- Sum / sum-of-squares of top-250 (fp64) ----------------
    double s = 0.0, s2 = 0.0;
    for (int c = tid; c < nchunk; c += nt) {
        float4 q = vals4[c];
        float fa[4] = { q.x, q.y, q.z, q.w };
#pragma unroll
        for (int j = 0; j < 4; ++j) {
            float v = clampf(fa[j]);
            if (key_of(v) > Kstar) {
                double d = (double)v;
                s  += d;
                s2 += d * d;
            }
        }
    }
    for (int i = tail0 + tid; i < V; i += nt) {
        float v = clampf(vals[i]);
        if (key_of(v) > Kstar) {
            double d = (double)v;
            s  += d;
            s2 += d * d;
        }
    }
    for (int off = 16; off > 0; off >>= 1) {   // wave32 reduction
        s  += __shfl_down(s,  off, 32);
        s2 += __shfl_down(s2, off, 32);
    }
    const int lane = tid & 31;
    if (lane == 0) { red_s[mywave] = s; red_s2[mywave] = s2; }
    __syncthreads();
    if (tid == 0) {
        double S = 0.0, S2 = 0.0;
        const int nw = nt >> 5;
        for (int w = 0; w < nw; ++w) { S += red_s[w]; S2 += red_s2[w]; }
        double tv = (double)val_of(Kstar);          // ties have identical values
        S  += (double)tieNeed * tv;
        S2 += (double)tieNeed * tv * tv;
        double mean = S / (double)K_TOP;
        double var  = (S2 - S * S / (double)K_TOP) / (double)(K_TOP - 1);  // ddof=1
        if (var < 0.0) var = 0.0;
        bc_mean = mean;
        bc_rsig = 1.0 / sqrt(var + VAR_EPS);
    }
    __syncthreads();
    const double mean = bc_mean;
    const double rsig = bc_rsig;

    // -------- Elementwise: exact GELU(z-1) * 0.4; scalar stride keeps global
    // -------- stores lane-coalesced (row stride is only 4B-aligned). --------
    float* __restrict__ orow = out + (size_t)row * (size_t)V;
    for (int i = tid; i < V; i += nt) {
        double v = (double)clampf(vals[i]);
        double z = (v - mean) * rsig - 1.0;
        double g = 0.5 * z * (1.0 + erf(z * 0.70710678118654752440));  // exact GELU
        orow[i] = (float)(g * OUT_SCALE);  // POWER=1 => signed power is identity
    }
}

extern "C" void kernel_launch(void* const* d_in, const int* in_sizes, int n_in,
                              void* d_out, int out_size, void* d_ws, size_t ws_size,
                              hipStream_t stream) {
    (void)n_in; (void)d_ws; (void)ws_size; (void)out_size;
    const float* in  = (const float*)d_in[0];
    float*       out = (float*)d_out;
    const int V = V_CLASSES;
    const int B = in_sizes[0] / V;                       // 2048 rows
    const size_t shmem = (size_t)V * sizeof(float);      // 87,364 B dynamic LDS
    (void)hipFuncSetAttribute((const void*)LNClampPowerScaleMap_kernel,
                              hipFuncAttributeMaxDynamicSharedMemorySize,
                              (int)shmem);
    LNClampPowerScaleMap_kernel<<<dim3(B), dim3(BLOCK_T), shmem, stream>>>(in, out, V);
}